// InvariantMessagePassingTP_45999099740407
// MI455X (gfx1250) — compile-verified
//
#include <hip/hip_runtime.h>

typedef __attribute__((ext_vector_type(2))) float v2f;
typedef __attribute__((ext_vector_type(8))) float v8f;

// L_MAP = [0, 1,1,1, 2,2,2,2,2, 3,3,3,3,3,3,3]
__device__ __forceinline__ int l_of_m(int m) {
    return (m == 0) ? 0 : (m < 4) ? 1 : (m < 9) ? 2 : 3;
}

// One wave (32 lanes) per receiver node. Per edge in the node's contiguous
// segment we issue 8x V_WMMA_F32_16X16X4_F32:
//   D(16 lm x 16 ch) += A(16 lm x 4 l) * B(4 l x 16 ch)
// with A the one-hot-scaled edge_attrs and B = sender_feats * tp_weights.
// Accumulation across edges stays in the WMMA C/D registers (no atomics).
// Segment bounds / node id / sender id are forced into SGPRs so the edge
// loop runs on SALU control flow with EXEC untouched (WMMA requirement).
__global__ __launch_bounds__(256) void imp_tp_wmma_kernel(
    const float* __restrict__ node_feats,   // [N,128]
    const float* __restrict__ edge_attrs,   // [E,16]
    const float* __restrict__ tp_weights,   // [E,4,128]
    const int*   __restrict__ sender_list,  // [E]
    const int*   __restrict__ first_occ,    // [N]
    float* __restrict__ out,                // [N,16,128]
    int nnodes, int nedges)
{
    const int wave = threadIdx.x >> 5;
    const int lane = threadIdx.x & 31;
    // Wave-uniform node id, forced scalar.
    const int node = __builtin_amdgcn_readfirstlane(blockIdx.x * 8 + wave);
    if (node >= nnodes) return;   // uniform per wave; active waves keep EXEC all-1s

    const int n    = lane & 15;   // A row (lm) for lower half / B column (channel)
    const int half = lane >> 4;   // 0 -> K rows {0,1}, 1 -> K rows {2,3}
    const int l0 = 2 * half;
    const int l1 = 2 * half + 1;

    // One-hot selectors for A[lm, l] = edge_attrs[lm] * (L_MAP[lm] == l)
    const int lm_of_row = l_of_m(n);
    const float ax_on = (lm_of_row == l0) ? 1.0f : 0.0f;
    const float ay_on = (lm_of_row == l1) ? 1.0f : 0.0f;

    // Scalar segment bounds -> SALU loop control.
    const int e0 = __builtin_amdgcn_readfirstlane(first_occ[node]);
    const int e1 = __builtin_amdgcn_readfirstlane(
        (node + 1 < nnodes) ? first_occ[node + 1] : nedges);

    v8f acc[8] = {};  // 8 channel tiles x (16 lm x 16 ch) f32 accumulators

    for (int e = e0; e < e1; ++e) {
        // A matrix (16x4): lane row n, K = {l0, l1} in {a.x, a.y}
        const float ea = edge_attrs[(size_t)e * 16 + n];
        v2f a;
        a.x = ea * ax_on;
        a.y = ea * ay_on;

        // Scalar sender id -> SGPR base address for the gather.
        const int s = __builtin_amdgcn_readfirstlane(sender_list[e]);
        const float* __restrict__ sf = node_feats + (size_t)s * 128;
        const float* __restrict__ w0 = tp_weights + ((size_t)e * 4 + l0) * 128;
        const float* __restrict__ w1 = tp_weights + ((size_t)e * 4 + l1) * 128;

        // Branch-free prefetch of the next edge's tp_weights block
        // (clamped index; speculative global_prefetch_b8, no EXEC churn).
        const int ep = (e + 1 < e1) ? (e + 1) : e;
        __builtin_prefetch(tp_weights + (size_t)ep * 4 * 128, 0, 0);

#pragma unroll
        for (int t = 0; t < 8; ++t) {
            const int c = t * 16 + n;
            const float sv = sf[c];
            v2f b;
            b.x = sv * w0[c];   // B[l0, c]
            b.y = sv * w1[c];   // B[l1, c]
            acc[t] = __builtin_amdgcn_wmma_f32_16x16x4_f32(
                /*neg_a=*/false, a, /*neg_b=*/false, b,
                /*c_mod=*/(short)0, acc[t],
                /*reuse_a=*/false, /*reuse_b=*/false);
        }
    }

    // D layout: VGPR r -> M=r (lanes 0-15) / M=r+8 (lanes 16-31), N = lane&15
    float* __restrict__ outn = out + (size_t)node * 16 * 128;
#pragma unroll
    for (int t = 0; t < 8; ++t) {
#pragma unroll
        for (int r = 0; r < 8; ++r) {
            outn[(size_t)(r + 8 * half) * 128 + t * 16 + n] = acc[t][r];
        }
    }
}

extern "C" void kernel_launch(void* const* d_in, const int* in_sizes, int n_in,
                              void* d_out, int out_size, void* d_ws, size_t ws_size,
                              hipStream_t stream) {
    const float* node_feats  = (const float*)d_in[0];  // [N,128]
    const float* edge_attrs  = (const float*)d_in[1];  // [E,16]
    const float* tp_weights  = (const float*)d_in[2];  // [E,4,128]
    const int*   sender_list = (const int*)d_in[3];    // [E]
    // d_in[4] = receiver_list: unused (segments come from first_occurences)
    const int*   first_occ   = (const int*)d_in[5];    // [N]
    float* out = (float*)d_out;                        // [N,16,128]

    const int nnodes = in_sizes[0] / 128;
    const int nedges = in_sizes[1] / 16;

    const int waves_per_block = 8;                     // 256 threads = 8 wave32
    const int blocks = (nnodes + waves_per_block - 1) / waves_per_block;

    imp_tp_wmma_kernel<<<blocks, 256, 0, stream>>>(
        node_feats, edge_attrs, tp_weights, sender_list, first_occ,
        out, nnodes, nedges);
}